// WaterNet2_76373108457552
// MI455X (gfx1250) — compile-verified
//
#include <hip/hip_runtime.h>
#include <hip/hip_bf16.h>

#define NT 365
#define NS 1024
#define NH 64
#define NG 32
#define NW 514          // NH*8 + 2
#define NWC 512         // columns we compute with WMMA (gate blocks 0..7)

typedef __attribute__((ext_vector_type(2))) float v2f;
typedef __attribute__((ext_vector_type(8))) float v8f;

// ---------------------------------------------------------------------------
// Kernel 1: W[s, n] = sum_g xc[s,g] * fc_w[n,g] + fc_b[n], n in [0, 512)
// One wave32 per 16x16 output tile, V_WMMA_F32_16X16X4_F32, K=32 -> 8 steps.
// ---------------------------------------------------------------------------
__global__ __launch_bounds__(256) void wn2_gemm_wmma(
    const float* __restrict__ xc,     // [NS, NG]
    const float* __restrict__ fc_w,   // [NW, NG]
    const float* __restrict__ fc_b,   // [NW]
    float* __restrict__ W)            // [NS, NWC]
{
    const int lane   = threadIdx.x & 31;
    const int wid    = (blockIdx.x * blockDim.x + threadIdx.x) >> 5; // 0..2047
    const int tile_m = wid >> 5;            // 0..63  (16 sites each)
    const int tile_n = wid & 31;            // 0..31  (16 cols each)
    const int half   = lane >> 4;           // 0 or 1 (selects K pair within chunk)
    const int idx16  = lane & 15;

    const int m = tile_m * 16 + idx16;      // A row  (site)
    const int n = tile_n * 16 + idx16;      // B col  (output feature)

    // C/D layout: VGPR v -> row = v + 8*half, col = idx16 (same for both halves)
    // -> bias depends only on col, uniform across the 8 accumulator VGPRs.
    const float bias = fc_b[n];
    v8f c = { bias, bias, bias, bias, bias, bias, bias, bias };

    #pragma unroll
    for (int kk = 0; kk < NG; kk += 4) {
        const int k0 = kk + 2 * half;
        v2f a, b;
        // A (16x4 f32): lanes 0-15 hold K = kk+0,kk+1 ; lanes 16-31 hold K = kk+2,kk+3
        a.x = xc[m * NG + k0];
        a.y = xc[m * NG + k0 + 1];
        // B (4x16 f32) = fc_w^T : B[k][n] = fc_w[n][k], mirrored layout
        b.x = fc_w[n * NG + k0];
        b.y = fc_w[n * NG + k0 + 1];
        c = __builtin_amdgcn_wmma_f32_16x16x4_f32(
                /*neg_a=*/false, a, /*neg_b=*/false, b,
                /*c_mod=*/(short)0, c, /*reuse_a=*/false, /*reuse_b=*/false);
    }

    const int col = tile_n * 16 + idx16;
    #pragma unroll
    for (int v = 0; v < 8; ++v) {
        const int row = tile_m * 16 + v + 8 * half;
        W[row * NWC + col] = c[v];
    }
}

// ---------------------------------------------------------------------------
// Kernel 2: qb[s] = relu(dot(xc[s,:], fc_w[513,:]) + fc_b[513]) / NH
// ---------------------------------------------------------------------------
__global__ __launch_bounds__(256) void wn2_qb(
    const float* __restrict__ xc,
    const float* __restrict__ fc_w,
    const float* __restrict__ fc_b,
    float* __restrict__ qb)
{
    const int s = blockIdx.x * blockDim.x + threadIdx.x;
    if (s >= NS) return;
    const float* wr = fc_w + (NW - 1) * NG;
    float acc = fc_b[NW - 1];
    #pragma unroll
    for (int g = 0; g < NG; ++g) acc += xc[s * NG + g] * wr[g];
    qb[s] = fmaxf(acc, 0.0f) * (1.0f / (float)NH);
}

// ---------------------------------------------------------------------------
// Kernel 3: gates (incl. softmax) + 365-step recurrent scan.
// One wave32 per site; lane l owns h = l and h = l + 32.
// All state lives in registers; per-step Y reduction via 5 shfl_xor hops.
// ---------------------------------------------------------------------------
__device__ __forceinline__ float wn2_sigmoid(float v) {
    return 1.0f / (1.0f + expf(-v));
}

__global__ __launch_bounds__(256) void wn2_scan(
    const float* __restrict__ x,      // [NT, NS, 4]
    const float* __restrict__ W,      // [NS, NWC]
    const float* __restrict__ qbuf,   // [NS]
    float* __restrict__ Y)            // [NT, NS]
{
    const int lane = threadIdx.x & 31;
    const int s    = (blockIdx.x * blockDim.x + threadIdx.x) >> 5;  // 0..1023
    const float* wr = W + s * NWC;

    // ---- load raw gate pre-activations: block j, h = lane and lane+32 ----
    float raw[8][2];
    #pragma unroll
    for (int j = 0; j < 8; ++j) {
        raw[j][0] = wr[j * NH + lane];
        raw[j][1] = wr[j * NH + lane + 32];
    }

    float gm[2], ge[2], go[2], gl[2], ga[2], gb[2], kb[2], gi[2];
    #pragma unroll
    for (int i = 0; i < 2; ++i) {
        gm[i] = expf(raw[0][i]) + 1.0f;
        ge[i] = wn2_sigmoid(raw[1][i]) * 2.0f;
        go[i] = wn2_sigmoid(raw[2][i]);
        gl[i] = expf(raw[3][i] * 2.0f);
        gb[i] = wn2_sigmoid(raw[5][i]);
        kb[i] = wn2_sigmoid(raw[6][i]) * 0.1f;
        gi[i] = wn2_sigmoid(raw[7][i]);
    }

    // ---- softmax over the 64 h values of block 4 (whole site = this wave) ----
    float mx = fmaxf(raw[4][0], raw[4][1]);
    #pragma unroll
    for (int off = 16; off > 0; off >>= 1)
        mx = fmaxf(mx, __shfl_xor(mx, off, 32));
    float e0 = expf(raw[4][0] - mx);
    float e1 = expf(raw[4][1] - mx);
    float sm = e0 + e1;
    #pragma unroll
    for (int off = 16; off > 0; off >>= 1)
        sm += __shfl_xor(sm, off, 32);
    const float inv = 1.0f / sm;
    ga[0] = e0 * inv;
    ga[1] = e1 * inv;

    const float qb = qbuf[s];

    // ---- recurrent scan: state in registers ----
    float S0[2] = {0.f, 0.f}, H0[2] = {0.f, 0.f}, G0[2] = {0.f, 0.f};
    const float4* x4 = reinterpret_cast<const float4*>(x);

    for (int t = 0; t < NT; ++t) {
        const float4 xv = x4[t * NS + s];   // broadcast: same addr for whole wave
        const float P = xv.x, E = xv.y, T1 = xv.z, T2 = xv.w;
        const float Ta = 0.5f * (T1 + T2);

        const bool valid = (T1 < 0.0f) && (T2 > 0.0f);
        const float denom = valid ? (T2 - T1) : 1.0f;
        float ratio = valid ? (T1 + T2) / denom : 0.0f;
        ratio = fminf(fmaxf(ratio, -0.999999f), 0.999999f);
        float rP = 1.0f - acosf(ratio) * (1.0f / 3.1415f);
        if (T1 >= 0.0f) rP = 1.0f;
        if (T2 <= 0.0f) rP = 0.0f;
        const float Ps = (1.0f - rP) * P;
        const float Pl = rP * P;

        float y = 0.0f;
        #pragma unroll
        for (int i = 0; i < 2; ++i) {
            const float S   = S0[i] + Ps;
            const float Sm  = fminf(S0[i], fmaxf(Ta * gm[i], 0.0f));
            const float H   = fmaxf(H0[i] + Sm + Pl * gi[i] - E * ge[i], 0.0f);
            const float Q1  = fmaxf(H - gl[i], 0.0f);
            const float Q2a = fminf(H, gl[i]) * go[i];
            const float Q2  = Q2a * (1.0f - gb[i]);
            const float G   = G0[i] + Q2a * gb[i];
            const float Q3  = G * kb[i];
            H0[i] = fminf(H - Q2a, gl[i]);
            G0[i] = G - Q3;
            S0[i] = S - Sm;
            y += (Q1 + Q2 + Q3 + qb) * ga[i];
        }
        #pragma unroll
        for (int off = 16; off > 0; off >>= 1)
            y += __shfl_xor(y, off, 32);
        if (lane == 0) Y[t * NS + s] = y;
    }
}

// ---------------------------------------------------------------------------
extern "C" void kernel_launch(void* const* d_in, const int* in_sizes, int n_in,
                              void* d_out, int out_size, void* d_ws, size_t ws_size,
                              hipStream_t stream) {
    (void)in_sizes; (void)n_in; (void)out_size; (void)ws_size;

    const float* x    = (const float*)d_in[0];   // [NT, NS, 4]
    const float* xc   = (const float*)d_in[1];   // [NS, NG]
    const float* fc_w = (const float*)d_in[2];   // [NW, NG]
    const float* fc_b = (const float*)d_in[3];   // [NW]
    float* Y = (float*)d_out;                    // [NT, NS]

    float* W  = (float*)d_ws;                    // [NS, NWC] = 2 MB
    float* qb = W + (size_t)NS * NWC;            // [NS]

    // 64 M-tiles x 32 N-tiles = 2048 waves = 256 blocks x 8 waves
    wn2_gemm_wmma<<<dim3(256), dim3(256), 0, stream>>>(xc, fc_w, fc_b, W);
    wn2_qb<<<dim3(4), dim3(256), 0, stream>>>(xc, fc_w, fc_b, qb);
    // 1024 sites = 1024 waves = 128 blocks x 8 waves
    wn2_scan<<<dim3(128), dim3(256), 0, stream>>>(x, W, qb, Y);
}